// Conv2d_80590766342288
// MI455X (gfx1250) — compile-verified
//
#include <hip/hip_runtime.h>

typedef __attribute__((ext_vector_type(2))) float v2f;
typedef __attribute__((ext_vector_type(8))) float v8f;
typedef int v4i_vs __attribute__((vector_size(4 * sizeof(int))));

#define GLOBAL_AS __attribute__((address_space(1)))
#define LDS_AS    __attribute__((address_space(3)))

#if __has_builtin(__builtin_amdgcn_global_load_async_to_lds_b128)
#define HAVE_ASYNC_LDS 1
#else
#define HAVE_ASYNC_LDS 0
#endif

namespace {
constexpr int CIN   = 32;
constexpr int COUT  = 32;
constexpr int H     = 256;
constexpr int W     = 256;
constexpr int KTOT  = CIN * 9;   // 288
constexpr int WTILE = 128;       // output columns per block
constexpr int COLS  = 144;       // LDS row stride (144*4B, 16B aligned; 144%64=16 -> lane halves
                                 // hit disjoint bank groups on B gathers)
}

__global__ __launch_bounds__(256)
void conv3x3_wmma_f32(const float* __restrict__ x,
                      const float* __restrict__ wgt,
                      const float* __restrict__ bias,
                      float* __restrict__ out) {
  __shared__ float xt[CIN * 3 * COLS];   // 55,296 B halo tile
  __shared__ float bs[COUT];

  const int tid   = threadIdx.x;
  const int b     = blockIdx.x;
  const int wseg  = b & 1;
  const int h     = (b >> 1) & (H - 1);
  const int n     = b >> 9;
  const int wbase = wseg * WTILE;

  // ---- zero the out-of-image row (only h==0 / h==H-1 blocks pay this) ----
  if (h == 0 || h == H - 1) {
    const int rz = (h == 0) ? 0 : 2;
    for (int idx = tid; idx < CIN * (COLS / 4); idx += 256) {
      const int ci = idx / (COLS / 4);
      const int s  = idx - ci * (COLS / 4);
      *reinterpret_cast<float4*>(&xt[(ci * 3 + rz) * COLS + s * 4]) =
          make_float4(0.f, 0.f, 0.f, 0.f);
    }
  }
  // ---- halo columns (left w-1, right w+WTILE), zero at image borders ----
  for (int idx = tid; idx < 192; idx += 256) {
    const int row = idx >> 1;
    const int e   = idx & 1;
    const int ci  = row / 3;
    const int r   = row - ci * 3;
    const int hr  = h - 1 + r;
    const int wc  = e ? (wbase + WTILE) : (wbase - 1);
    float v = 0.f;
    if (hr >= 0 && hr < H && wc >= 0 && wc < W)
      v = x[(((size_t)n * CIN + ci) * H + hr) * W + wc];
    xt[row * COLS + (e ? (4 + WTILE) : 3)] = v;
  }
  if (tid < COUT) bs[tid] = bias[tid];

  // ---- interior: async global->LDS (ASYNCcnt) with sync fallback ----
  for (int idx = tid; idx < 96 * (WTILE / 4); idx += 256) {
    const int row = idx >> 5;            // (ci*3 + r)
    const int seg = idx & 31;
    const int ci  = row / 3;
    const int r   = row - ci * 3;
    const int hr  = h - 1 + r;
    if (hr >= 0 && hr < H) {
      const float* src = x + ((((size_t)n * CIN + ci) * H + hr) * W + wbase) + seg * 4;
      float* dst = &xt[row * COLS + 4 + seg * 4];          // 16B aligned
#if HAVE_ASYNC_LDS
      __builtin_amdgcn_global_load_async_to_lds_b128(
          (GLOBAL_AS v4i_vs*)src, (LDS_AS v4i_vs*)dst, 0, 0);
#else
      *reinterpret_cast<float4*>(dst) = *reinterpret_cast<const float4*>(src);
#endif
    }
  }
#if HAVE_ASYNC_LDS
#if __has_builtin(__builtin_amdgcn_s_wait_asynccnt)
  __builtin_amdgcn_s_wait_asynccnt(0);
#else
  asm volatile("s_wait_asynccnt 0" ::: "memory");
#endif
#endif
  __syncthreads();

  // ---- implicit GEMM, K permuted into LDS-contiguous chunks ----
  // WMMA f32 16x16x4 fragments: lanes 0-15 hold K{0,1}, lanes 16-31 hold K{2,3}.
  const int wave   = tid >> 5;
  const int lane   = tid & 31;
  const int lane16 = lane & 15;
  const int hi     = lane >> 4;
  const int mtile  = wave & 1;       // co base 0 / 16
  const int nt0    = wave >> 1;      // column tiles nt0, nt0+4
  const int j0     = nt0 * 16 + lane16;

  // A (weights, OIHW == co*288 + row*3 + kw): per-lane bases, all loads at immediates.
  const float* aA = wgt + (size_t)(mtile * 16 + lane16) * KTOT + 3 * hi;       // kw{0,1} chunks
  const float* aB = wgt + (size_t)(mtile * 16 + lane16) * KTOT + 2 + 6 * hi;   // kw=2 chunks
  // B (im2col from xt): per-lane bases, all loads at immediates.
  const float* bA0 = &xt[hi * COLS + 3 + j0];
  const float* bA1 = bA0 + 64;
  const float* bB0 = &xt[2 * hi * COLS + 5 + j0];
  const float* bB1 = bB0 + 64;

  v8f acc0 = {};
  v8f acc1 = {};

  // Type-A chunks: K = [(2a,kw0),(2a,kw1),(2a+1,kw0),(2a+1,kw1)], a = 0..47
  #pragma unroll 8
  for (int a = 0; a < 48; ++a) {
    v2f av;  av.x = aA[6 * a];          av.y = aA[6 * a + 1];
    const int e = 2 * COLS * a;
    v2f b0;  b0.x = bA0[e];             b0.y = bA0[e + 1];
    v2f b1;  b1.x = bA1[e];             b1.y = bA1[e + 1];
    acc0 = __builtin_amdgcn_wmma_f32_16x16x4_f32(false, av, false, b0,
                                                 (short)0, acc0, false, false);
    acc1 = __builtin_amdgcn_wmma_f32_16x16x4_f32(false, av, false, b1,
                                                 (short)0, acc1, false, false);
  }
  // Type-B chunks: K = [(4c,2),(4c+1,2),(4c+2,2),(4c+3,2)], c = 0..23
  #pragma unroll 8
  for (int c = 0; c < 24; ++c) {
    v2f av;  av.x = aB[12 * c];         av.y = aB[12 * c + 3];
    const int e = 4 * COLS * c;
    v2f b0;  b0.x = bB0[e];             b0.y = bB0[e + COLS];
    v2f b1;  b1.x = bB1[e];             b1.y = bB1[e + COLS];
    acc0 = __builtin_amdgcn_wmma_f32_16x16x4_f32(false, av, false, b0,
                                                 (short)0, acc0, false, false);
    acc1 = __builtin_amdgcn_wmma_f32_16x16x4_f32(false, av, false, b1,
                                                 (short)0, acc1, false, false);
  }

  // ---- epilogue: bias + coalesced store ----
  // C/D: VGPR g holds M = g (lanes 0-15) / g+8 (lanes 16-31), N = lane%16
  #pragma unroll
  for (int g = 0; g < 8; ++g) {
    const int co = mtile * 16 + hi * 8 + g;
    const float bb = bs[co];
    float* op = out + (((size_t)n * COUT + co) * H + h) * W + wbase;
    op[j0]      = acc0[g] + bb;
    op[j0 + 64] = acc1[g] + bb;
  }
}

extern "C" void kernel_launch(void* const* d_in, const int* in_sizes, int n_in,
                              void* d_out, int out_size, void* d_ws, size_t ws_size,
                              hipStream_t stream) {
  const float* x   = (const float*)d_in[0];   // (16, 32, 256, 256) fp32
  const float* wgt = (const float*)d_in[1];   // (32, 32, 3, 3)     fp32
  const float* bia = (const float*)d_in[2];   // (32,)              fp32
  float* out = (float*)d_out;                 // (16, 32, 256, 256) fp32

  const int nblocks = 16 * 256 * 2;           // (n, h, w-half)
  conv3x3_wmma_f32<<<dim3(nblocks), dim3(256), 0, stream>>>(x, wgt, bia, out);
}